// MultiModalLightGCN_84550726189741
// MI455X (gfx1250) — compile-verified
//
#include <hip/hip_runtime.h>
#include <hip/hip_bf16.h>

#define NUu   80000
#define MIi   40000
#define Dm    64
#define Hh    8
#define LL    2
#define NLAY  3
#define Bb    512
#define Nn    (NUu + MIi)

typedef float v2f __attribute__((ext_vector_type(2)));
typedef float v8f __attribute__((ext_vector_type(8)));
typedef int v4i_vs __attribute__((vector_size(16)));  // matches builtin prototype

// ---------------- helpers ----------------
__device__ __forceinline__ float lrelu2(float x) { return x > 0.f ? x : 0.2f * x; }

// order-preserving float->uint encoding for atomic max (handles negatives)
__device__ __forceinline__ unsigned fOrd(float x) {
    unsigned u = __float_as_uint(x);
    return (u & 0x80000000u) ? ~u : (u | 0x80000000u);
}
__device__ __forceinline__ float fUnord(unsigned u) {
    return (u & 0x80000000u) ? __uint_as_float(u ^ 0x80000000u) : __uint_as_float(~u);
}

// fp32 WMMA 16x16 tile, K-step 4.  A: lane=M (0..15), slots = K(ks),K(ks+1), ks = lane<16?0:2.
// B: lane=N, same K slots.  C/D: 8 regs, row = r + (lane<16?0:8), col = lane&15.
__device__ __forceinline__ v8f wmma_f32_16x16x4(v2f a, v2f b, v8f c) {
#if __has_builtin(__builtin_amdgcn_wmma_f32_16x16x4_f32)
    return __builtin_amdgcn_wmma_f32_16x16x4_f32(false, a, false, b, (short)0, c, false, false);
#else
    int lane = (int)(threadIdx.x & 31);
    int col = lane & 15;
    int mo = (lane < 16) ? 0 : 8;
#pragma unroll
    for (int r = 0; r < 8; ++r) {
        int row = mo + r;
        float s = c[r];
#pragma unroll
        for (int kk = 0; kk < 4; ++kk) {
            float av = __shfl(a[kk & 1], ((kk < 2) ? 0 : 16) + row, 32);
            float bv = __shfl(b[kk & 1], ((kk < 2) ? 0 : 16) + col, 32);
            s += av * bv;
        }
        c[r] = s;
    }
    return c;
#endif
}

// 16-byte global -> LDS copy through the gfx1250 async engine (ASYNCcnt-tracked),
// with a plain register-copy fallback if the builtins are not declared.
__device__ __forceinline__ void g2lds16(const float* g, float* l) {
#if __has_builtin(__builtin_amdgcn_global_load_async_to_lds_b128)
    typedef __attribute__((address_space(1))) v4i_vs* gp_t;
    typedef __attribute__((address_space(3))) v4i_vs* lp_t;
    __builtin_amdgcn_global_load_async_to_lds_b128((gp_t)(void*)g, (lp_t)(void*)l, 0, 0);
#else
    float4 t = *(const float4*)g;
    *(float4*)l = t;
#endif
}
__device__ __forceinline__ void g2lds_wait() {
#if __has_builtin(__builtin_amdgcn_global_load_async_to_lds_b128) && \
    __has_builtin(__builtin_amdgcn_s_wait_asynccnt)
    __builtin_amdgcn_s_wait_asynccnt(0);
#endif
}

// ---------------- elementwise ----------------
__global__ void k_fill0(float* p, size_t n) {
    for (size_t i = (size_t)blockIdx.x * blockDim.x + threadIdx.x; i < n;
         i += (size_t)gridDim.x * blockDim.x)
        p[i] = 0.f;
}

__global__ void k_init_cur(const float* ue, const float* ie, float* cur, float* acc) {
    size_t n = (size_t)Nn * Dm;
    size_t nu = (size_t)NUu * Dm;
    for (size_t i = (size_t)blockIdx.x * blockDim.x + threadIdx.x; i < n;
         i += (size_t)gridDim.x * blockDim.x) {
        float v = (i < nu) ? ue[i] : ie[i - nu];
        cur[i] = v;
        acc[i] = v;
    }
}

__global__ void k_add(float* acc, const float* x, size_t n) {
    for (size_t i = (size_t)blockIdx.x * blockDim.x + threadIdx.x; i < n;
         i += (size_t)gridDim.x * blockDim.x)
        acc[i] += x[i];
}

__global__ void k_scale(float* p, float s, size_t n) {
    for (size_t i = (size_t)blockIdx.x * blockDim.x + threadIdx.x; i < n;
         i += (size_t)gridDim.x * blockDim.x)
        p[i] *= s;
}

// ---------------- LightGCN SpMM: wave per edge, lane covers d and d+32 ----------------
// Latency-bound gather (x is L2-resident): software-prefetch the next edge's source row,
// lanes spread over the 256B row so the whole line set is requested.
__global__ void k_spmm(const int* __restrict__ row, const int* __restrict__ col,
                       const float* __restrict__ val, const float* __restrict__ x,
                       float* __restrict__ y, int E) {
    int wid = (int)((blockIdx.x * blockDim.x + threadIdx.x) >> 5);
    int lane = (int)(threadIdx.x & 31);
    int nw = (int)((gridDim.x * blockDim.x) >> 5);
    for (int e = wid; e < E; e += nw) {
        int en = e + nw;
        if (en < E)
            __builtin_prefetch(x + (size_t)col[en] * Dm + lane * 2, 0, 3);
        int r = row[e], c = col[e];
        float v = val[e];
        const float* xc = x + (size_t)c * Dm;
        float* yr = y + (size_t)r * Dm;
        atomicAdd(&yr[lane], v * xc[lane]);
        atomicAdd(&yr[lane + 32], v * xc[lane + 32]);
    }
}

// ---------------- generic GEMM  Y[M x 64] = X[M x K] @ W[K x 64] (+bias) ----------------
__global__ void k_gemm_n64(const float* __restrict__ X, const float* __restrict__ W,
                           const float* __restrict__ bias, float* __restrict__ Y,
                           int M, int K) {
    int wave = (int)((blockIdx.x * blockDim.x + threadIdx.x) >> 5);
    int lane = (int)(threadIdx.x & 31);
    int nTiles = (M >> 4) * 4;
    if (wave >= nTiles) return;
    int m0 = (wave >> 2) * 16, n0 = (wave & 3) * 16;
    int mrow = m0 + (lane & 15);
    int nc = n0 + (lane & 15);
    int ks = (lane < 16) ? 0 : 2;
    const float* xr = X + (size_t)mrow * K + ks;
    v8f c = {};
    for (int k = 0; k < K; k += 4) {
        v2f a, b;
        a[0] = xr[k];
        a[1] = xr[k + 1];
        b[0] = W[(size_t)(k + ks) * Dm + nc];
        b[1] = W[(size_t)(k + ks + 1) * Dm + nc];
        c = wmma_f32_16x16x4(a, b, c);
    }
    int mo = (lane < 16) ? 0 : 8;
    float bv = bias ? bias[nc] : 0.f;
    float* yr = Y + (size_t)(m0 + mo) * Dm + nc;
#pragma unroll
    for (int r = 0; r < 8; ++r) yr[(size_t)r * Dm] = c[r] + bv;
}

// ---------------- G += mm.T @ st  (64x64 over K=M rows, split in K-chunks) ----------------
__global__ void k_gemmT64(const float* __restrict__ A, const float* __restrict__ Bm,
                          float* __restrict__ G, int M) {
    // blockDim = 512 (16 waves), wave w -> tile (w>>2, w&3); gridDim.x K-chunks
    int w = (int)(threadIdx.x >> 5);
    int lane = (int)(threadIdx.x & 31);
    int m0 = (w >> 2) * 16, n0 = (w & 3) * 16;
    int ks = (lane < 16) ? 0 : 2;
    int mi = m0 + (lane & 15);
    int ni = n0 + (lane & 15);
    int kpc = M / gridDim.x;  // caller guarantees divisible by 4
    int k0 = blockIdx.x * kpc;
    int k1 = k0 + kpc;
    v8f c = {};
    for (int k = k0; k < k1; k += 4) {
        v2f a, b;
        a[0] = A[(size_t)(k + ks) * Dm + mi];
        a[1] = A[(size_t)(k + ks + 1) * Dm + mi];
        b[0] = Bm[(size_t)(k + ks) * Dm + ni];
        b[1] = Bm[(size_t)(k + ks + 1) * Dm + ni];
        c = wmma_f32_16x16x4(a, b, c);
    }
    int mo = (lane < 16) ? 0 : 8;
#pragma unroll
    for (int r = 0; r < 8; ++r) atomicAdd(&G[(size_t)(m0 + mo + r) * Dm + ni], c[r]);
}

// ---------------- GAT pieces ----------------
__global__ void k_el_er(const float* __restrict__ h, const float* __restrict__ al,
                        const float* __restrict__ ar, float* __restrict__ el,
                        float* __restrict__ er, int M) {
    int n = M * Hh;
    for (int idx = blockIdx.x * blockDim.x + threadIdx.x; idx < n;
         idx += gridDim.x * blockDim.x) {
        int node = idx >> 3, hh = idx & 7;
        const float* hp = h + (size_t)node * Dm + hh * 8;
        const float* alp = al + hh * 8;
        const float* arp = ar + hh * 8;
        float sl = 0.f, sr = 0.f;
#pragma unroll
        for (int d = 0; d < 8; ++d) {
            sl += hp[d] * alp[d];
            sr += hp[d] * arp[d];
        }
        el[idx] = sl;
        er[idx] = sr;
    }
}

__global__ void k_edge_max(const int* __restrict__ src, const int* __restrict__ dst,
                           const float* __restrict__ el, const float* __restrict__ er,
                           unsigned* __restrict__ mmax, int E) {
    for (int e = blockIdx.x * blockDim.x + threadIdx.x; e < E; e += gridDim.x * blockDim.x) {
        int s = src[e], d = dst[e];
#pragma unroll
        for (int h = 0; h < Hh; ++h) {
            float ee = lrelu2(el[s * Hh + h] + er[d * Hh + h]);
            atomicMax(&mmax[d * Hh + h], fOrd(ee));
        }
    }
}

__global__ void k_edge_sum(const int* __restrict__ src, const int* __restrict__ dst,
                           const float* __restrict__ el, const float* __restrict__ er,
                           const unsigned* __restrict__ mmax, float* __restrict__ ssum, int E) {
    for (int e = blockIdx.x * blockDim.x + threadIdx.x; e < E; e += gridDim.x * blockDim.x) {
        int s = src[e], d = dst[e];
#pragma unroll
        for (int h = 0; h < Hh; ++h) {
            float ee = lrelu2(el[s * Hh + h] + er[d * Hh + h]);
            atomicAdd(&ssum[d * Hh + h], expf(ee - fUnord(mmax[d * Hh + h])));
        }
    }
}

__global__ void k_edge_acc(const int* __restrict__ src, const int* __restrict__ dst,
                           const float* __restrict__ el, const float* __restrict__ er,
                           const unsigned* __restrict__ mmax, const float* __restrict__ ssum,
                           const float* __restrict__ h, float* __restrict__ out, int E) {
    for (int e = blockIdx.x * blockDim.x + threadIdx.x; e < E; e += gridDim.x * blockDim.x) {
        int s = src[e], d = dst[e];
#pragma unroll
        for (int hh = 0; hh < Hh; ++hh) {
            float ee = lrelu2(el[s * Hh + hh] + er[d * Hh + hh]);
            float wgt = expf(ee - fUnord(mmax[d * Hh + hh])) / ssum[d * Hh + hh];
            const float* hs = h + (size_t)s * Dm + hh * 8;
            float* od = out + (size_t)d * Dm + hh * 8;
#pragma unroll
            for (int dd = 0; dd < 8; ++dd) atomicAdd(&od[dd], wgt * hs[dd]);
        }
    }
}

__global__ void k_bias_lrelu(const float* __restrict__ in, const float* __restrict__ bias,
                             float* __restrict__ out, int M) {
    size_t n = (size_t)M * Dm;
    for (size_t i = (size_t)blockIdx.x * blockDim.x + threadIdx.x; i < n;
         i += (size_t)gridDim.x * blockDim.x)
        out[i] = lrelu2(in[i] + bias[i & 63]);
}

// ---------------- OT cost pieces ----------------
__global__ void k_colms(const float* __restrict__ x, float* __restrict__ out, int M) {
    __shared__ float sm[256];
    int d = (int)(threadIdx.x & 63), sub = (int)(threadIdx.x >> 6);
    float acc = 0.f;
    for (int r = blockIdx.x * 4 + sub; r < M; r += gridDim.x * 4) {
        float v = x[(size_t)r * Dm + d];
        acc += v * v;
    }
    sm[threadIdx.x] = acc;
    __syncthreads();
    if (threadIdx.x < 64) {
        float s = sm[d] + sm[d + 64] + sm[d + 128] + sm[d + 192];
        atomicAdd(&out[d], s);
    }
}

__global__ void k_build_C(const float* __restrict__ colms, float* __restrict__ C) {
    int idx = blockIdx.x * blockDim.x + threadIdx.x;
    if (idx < Dm * Dm) {
        int i = idx >> 6, j = idx & 63;
        float g = C[idx];
        const float invM = 1.f / (float)MIi;
        C[idx] = (colms[i] * invM + colms[64 + j] * invM - 2.f * g * invM) * 100.f;
    }
}

__global__ void k_sinkhorn(const float* __restrict__ C, const float* __restrict__ delta,
                           float* __restrict__ Tp) {
    __shared__ float Ks[Dm * Dm];
    __shared__ float u[Dm], v[Dm];
    int t = (int)threadIdx.x;  // blockDim = 64
    for (int i = t; i < Dm * Dm; i += 64) Ks[i] = expf(-C[i] * 100.f);  // eps = 0.01
    u[t] = 1.f / 64.f;
    v[t] = 1.f / 64.f;
    __syncthreads();
    for (int it = 0; it < 100; ++it) {
        float s = 0.f;
        for (int j = 0; j < Dm; ++j) s += Ks[t * Dm + j] * v[j];
        __syncthreads();
        u[t] = (1.f / 64.f) / (s + 1e-8f);
        __syncthreads();
        float s2 = 0.f;
        for (int i = 0; i < Dm; ++i) s2 += Ks[i * Dm + t] * u[i];
        __syncthreads();
        v[t] = (1.f / 64.f) / (s2 + 1e-8f);
        __syncthreads();
    }
    for (int i = t; i < Dm * Dm; i += 64) {
        int r = i >> 6, c2 = i & 63;
        Tp[i] = u[r] * Ks[i] * v[c2] + delta[i];
    }
}

// ---------------- combine + LayerNorm + L2 (wave per row) ----------------
__global__ void k_combine_ln(const float* __restrict__ zcf, const float* __restrict__ aimg,
                             const float* __restrict__ atxt, const float* __restrict__ alpha_p,
                             const float* __restrict__ beta_p, const float* __restrict__ g,
                             const float* __restrict__ b, float* __restrict__ z, int M) {
    int wid = (int)((blockIdx.x * blockDim.x + threadIdx.x) >> 5);
    int lane = (int)(threadIdx.x & 31);
    int nw = (int)((gridDim.x * blockDim.x) >> 5);
    float alpha = alpha_p[0], beta = beta_p[0];
    float w0 = 1.f - alpha - beta;
    for (int r = wid; r < M; r += nw) {
        size_t base = (size_t)r * Dm;
        float z0 = w0 * zcf[base + lane] + alpha * aimg[base + lane] + beta * atxt[base + lane];
        float z1 = w0 * zcf[base + lane + 32] + alpha * aimg[base + lane + 32] +
                   beta * atxt[base + lane + 32];
        float s = z0 + z1;
        for (int o = 16; o > 0; o >>= 1) s += __shfl_xor(s, o, 32);
        float mu = s * (1.f / 64.f);
        float d0 = z0 - mu, d1 = z1 - mu;
        float vv = d0 * d0 + d1 * d1;
        for (int o = 16; o > 0; o >>= 1) vv += __shfl_xor(vv, o, 32);
        float inv = 1.f / sqrtf(vv * (1.f / 64.f) + 1e-5f);
        float y0 = d0 * inv * g[lane] + b[lane];
        float y1 = d1 * inv * g[lane + 32] + b[lane + 32];
        float ss = y0 * y0 + y1 * y1;
        for (int o = 16; o > 0; o >>= 1) ss += __shfl_xor(ss, o, 32);
        float nrm = fmaxf(sqrtf(ss), 1e-12f);
        z[base + lane] = y0 / nrm;
        z[base + lane + 32] = y1 / nrm;
    }
}

__global__ void k_gather(const float* __restrict__ zu, const int* __restrict__ users,
                         float* __restrict__ zug) {
    int i = blockIdx.x * blockDim.x + threadIdx.x;
    if (i < Bb * Dm) {
        int u = users[i >> 6];
        zug[i] = zu[(size_t)u * Dm + (i & 63)];
    }
}

// ---------------- final scores: OUT[512 x 40000] = ZU @ Z^T ----------------
// Block owns one 16-item N-tile: stage the 16x64 Z tile in LDS once (async copy engine),
// then 8 waves cover all 32 M-tiles from LDS -> 32x reuse of the staged tile.
__global__ void k_score_lds(const float* __restrict__ ZU, const float* __restrict__ Z,
                            float* __restrict__ OUT) {
    __shared__ float zt[16 * Dm];  // 4 KB
    int n0 = blockIdx.x * 16;
    int tid = (int)threadIdx.x;  // 256
    g2lds16(Z + (size_t)n0 * Dm + tid * 4, &zt[tid * 4]);  // 256 x 16B = 4 KB
    g2lds_wait();
    __syncthreads();

    int w = tid >> 5;
    int lane = tid & 31;
    int ni = lane & 15;
    int ks = (lane < 16) ? 0 : 2;
    int mo = (lane < 16) ? 0 : 8;
#pragma unroll
    for (int t = 0; t < 4; ++t) {
        int m0 = (w + t * 8) * 16;
        int mi = m0 + (lane & 15);
        const float* ar = ZU + (size_t)mi * Dm + ks;
        const float* br = &zt[ni * Dm + ks];
        v8f c = {};
#pragma unroll 4
        for (int k = 0; k < Dm; k += 4) {
            v2f a, b;
            a[0] = ar[k];
            a[1] = ar[k + 1];
            b[0] = br[k];
            b[1] = br[k + 1];
            c = wmma_f32_16x16x4(a, b, c);
        }
#pragma unroll
        for (int r = 0; r < 8; ++r) OUT[(size_t)(m0 + mo + r) * MIi + n0 + ni] = c[r];
    }
}

// ---------------- driver ----------------
extern "C" void kernel_launch(void* const* d_in, const int* in_sizes, int n_in,
                              void* d_out, int out_size, void* d_ws, size_t ws_size,
                              hipStream_t stream) {
    const float* user_emb = (const float*)d_in[0];
    const float* item_emb = (const float*)d_in[1];
    const int* g_row = (const int*)d_in[2];
    const int* g_col = (const int*)d_in[3];
    const float* g_val = (const float*)d_in[4];
    const float* txt_feat = (const float*)d_in[5];
    const float* img_feat = (const float*)d_in[6];
    const int* t_src = (const int*)d_in[7];
    const int* t_dst = (const int*)d_in[8];
    const int* i_src = (const int*)d_in[9];
    const int* i_dst = (const int*)d_in[10];
    const float* txt_proj_w = (const float*)d_in[11];
    const float* txt_proj_b = (const float*)d_in[12];
    const float* txt_fc_w = (const float*)d_in[13];
    const float* txt_attn_l = (const float*)d_in[14];
    const float* txt_attn_r = (const float*)d_in[15];
    const float* txt_bias = (const float*)d_in[16];
    const float* img_proj_w = (const float*)d_in[17];
    const float* img_proj_b = (const float*)d_in[18];
    const float* img_fc_w = (const float*)d_in[19];
    const float* img_attn_l = (const float*)d_in[20];
    const float* img_attn_r = (const float*)d_in[21];
    const float* img_bias = (const float*)d_in[22];
    const float* delta_img = (const float*)d_in[23];
    const float* delta_txt = (const float*)d_in[24];
    const float* alpha_p = (const float*)d_in[25];
    const float* beta_p = (const float*)d_in[26];
    const float* ln_g = (const float*)d_in[27];
    const float* ln_b = (const float*)d_in[28];
    const int* users = (const int*)d_in[29];

    const int EG = in_sizes[2];  // 3,200,000
    const int EB = in_sizes[7];  // 440,000

    const size_t NN64 = (size_t)Nn * Dm;   // 7,680,000
    const size_t MI64 = (size_t)MIi * Dm;  // 2,560,000

    float* ws = (float*)d_ws;
    float* acc = ws;           // [NN64] -> becomes light (scaled 1/4)
    float* scr = ws + NN64;    // 2*NN64 scratch union
    float* cur = scr;
    float* nxt = scr + NN64;
    // GAT branch scratch (reuses cur/nxt region after phase 1)
    float* xbuf = scr;
    float* hbuf = scr + MI64;
    float* obuf = scr + 2 * MI64;
    float* el = scr + 3 * MI64;
    float* er = el + (size_t)MIi * Hh;
    unsigned* mmax = (unsigned*)(er + (size_t)MIi * Hh);
    float* ssum = (float*)mmax + (size_t)MIi * Hh;
    float* aimg = scr + 4 * MI64;
    float* atxt = scr + 5 * MI64;
    float* zbuf = scr;  // reuses xbuf region (dead when combine runs)
    float* zi_txt = ws + 3 * NN64;
    float* zi_img = zi_txt + MI64;
    float* small = zi_img + MI64;
    float* colms = small;          // 128
    float* Cmat = small + 128;     // 4096
    float* Tp = Cmat + 4096;       // 4096
    float* zug = Tp + 4096;        // 32768
    float* zcf = acc + (size_t)NUu * Dm;

    const int TPB = 256;
    const int gemmBlocks = ((MIi / 16) * 4 * 32 + TPB - 1) / TPB;  // 1250

    // ---- Phase 1: LightGCN ----
    k_init_cur<<<4096, TPB, 0, stream>>>(user_emb, item_emb, cur, acc);
    for (int l = 0; l < NLAY; ++l) {
        k_fill0<<<4096, TPB, 0, stream>>>(nxt, NN64);
        k_spmm<<<8192, TPB, 0, stream>>>(g_row, g_col, g_val, cur, nxt, EG);
        k_add<<<4096, TPB, 0, stream>>>(acc, nxt, NN64);
        float* t = cur; cur = nxt; nxt = t;
    }
    k_scale<<<4096, TPB, 0, stream>>>(acc, 0.25f, NN64);

    // ---- Phase 2: GAT branches ----
    auto run_branch = [&](const float* feat, int K, const int* src, const int* dst,
                          const float* pw, const float* pb, const float* fcw,
                          const float* al, const float* ar, const float* bias, float* zi) {
        k_gemm_n64<<<gemmBlocks, TPB, 0, stream>>>(feat, pw, pb, xbuf, MIi, K);
        for (int l = 0; l < LL; ++l) {
            k_gemm_n64<<<gemmBlocks, TPB, 0, stream>>>(xbuf, fcw + l * Dm * Dm, nullptr,
                                                       hbuf, MIi, Dm);
            k_el_er<<<1024, TPB, 0, stream>>>(hbuf, al + l * Dm, ar + l * Dm, el, er, MIi);
            k_fill0<<<1024, TPB, 0, stream>>>((float*)mmax, (size_t)MIi * Hh);
            k_fill0<<<1024, TPB, 0, stream>>>(ssum, (size_t)MIi * Hh);
            k_fill0<<<4096, TPB, 0, stream>>>(obuf, MI64);
            k_edge_max<<<2048, TPB, 0, stream>>>(src, dst, el, er, mmax, EB);
            k_edge_sum<<<2048, TPB, 0, stream>>>(src, dst, el, er, mmax, ssum, EB);
            k_edge_acc<<<2048, TPB, 0, stream>>>(src, dst, el, er, mmax, ssum, hbuf, obuf, EB);
            float* dbuf = (l == LL - 1) ? zi : xbuf;
            k_bias_lrelu<<<4096, TPB, 0, stream>>>(obuf, bias + l * Dm, dbuf, MIi);
        }
    };
    run_branch(txt_feat, 768, t_src, t_dst, txt_proj_w, txt_proj_b, txt_fc_w,
               txt_attn_l, txt_attn_r, txt_bias, zi_txt);
    run_branch(img_feat, 512, i_src, i_dst, img_proj_w, img_proj_b, img_fc_w,
               img_attn_l, img_attn_r, img_bias, zi_img);

    // ---- Phase 3: OT alignment ----
    auto run_ot = [&](const float* mm, const float* delta, float* aout) {
        k_fill0<<<1, 128, 0, stream>>>(colms, 128);
        k_fill0<<<16, TPB, 0, stream>>>(Cmat, (size_t)Dm * Dm);
        k_colms<<<128, TPB, 0, stream>>>(mm, colms, MIi);
        k_colms<<<128, TPB, 0, stream>>>(zcf, colms + 64, MIi);
        k_gemmT64<<<100, 512, 0, stream>>>(mm, zcf, Cmat, MIi);  // 100 chunks of 400 rows
        k_build_C<<<16, TPB, 0, stream>>>(colms, Cmat);
        k_sinkhorn<<<1, 64, 0, stream>>>(Cmat, delta, Tp);
        k_gemm_n64<<<gemmBlocks, TPB, 0, stream>>>(mm, Tp, nullptr, aout, MIi, Dm);
    };
    run_ot(zi_img, delta_img, aimg);
    run_ot(zi_txt, delta_txt, atxt);

    // ---- Phase 4: combine + LN + L2, gather, scores ----
    k_combine_ln<<<1250, TPB, 0, stream>>>(zcf, aimg, atxt, alpha_p, beta_p, ln_g, ln_b,
                                           zbuf, MIi);
    k_gather<<<(Bb * Dm + TPB - 1) / TPB, TPB, 0, stream>>>(acc, users, zug);
    k_score_lds<<<MIi / 16, TPB, 0, stream>>>(zug, zbuf, (float*)d_out);
}